// SimpleGCN_42417097015621
// MI455X (gfx1250) — compile-verified
//
#include <hip/hip_runtime.h>
#include <hip/hip_bf16.h>

// ---------------------------------------------------------------------------
// Two-layer GCN (PyG GCNConv semantics) for MI455X / gfx1250.
//   deg = 1 + in-degree ; dinv = rsqrt(deg)
//   h   = relu( scatter(dinv[s]*dinv[d] * (xW1)[s] -> d) + dinv^2*(xW1) + b1 )
//   z   =       scatter(dinv[s]*dinv[d] * (hW2)[s] -> d) + dinv^2*(hW2) + b2
// GEMMs run on the WMMA pipe in full fp32 (v_wmma_f32_16x16x4_f32) and fuse
// the self-loop + bias accumulator init into their epilogue.
// Edge phase is L2-resident (h fits in 192MB L2) -> coalesced wave gathers
// plus global_atomic_add_f32 scatters.
// ---------------------------------------------------------------------------

typedef __attribute__((ext_vector_type(2))) float v2f;
typedef __attribute__((ext_vector_type(8))) float v8f;

#define N_FEAT_IN 128

// ----------------------------- degree kernels ------------------------------

__global__ void k_init_deg(float* deg, int n) {
    int i = blockIdx.x * blockDim.x + threadIdx.x;
    if (i < n) deg[i] = 1.0f;  // self loop
}

__global__ void k_count_deg(const int* __restrict__ dst, float* deg, int e) {
    int i = blockIdx.x * blockDim.x + threadIdx.x;
    if (i < e) atomicAdd(&deg[dst[i]], 1.0f);
}

__global__ void k_dinv(const float* __restrict__ deg, float* dinv, int n) {
    int i = blockIdx.x * blockDim.x + threadIdx.x;
    if (i < n) dinv[i] = rsqrtf(deg[i]);  // deg >= 1 always
}

// ----------------------------- WMMA GEMM -----------------------------------
// H[n, OUT]   = X[n, K] @ W[K, OUT]                       (edge-gather source)
// AGG[n, OUT] = bias + dinv[n]^2 * H[n, OUT]              (fused accumulator
//                                                          init: self-loop+b)
// One block per 16-row tile; one wave per 16-column tile (OUT/16 waves).
// A (16xK) tile staged in LDS; one v_wmma_f32_16x16x4_f32 per K-step of 4.
//
// fp32 16x16x4 operand layout (ISA 7.12.2):
//   A 16x4 : lane m (0..15) vgpr0 = A[m][0], vgpr1 = A[m][1]
//            lane m+16      vgpr0 = A[m][2], vgpr1 = A[m][3]
//   B 4x16 : lane n (0..15) vgpr0 = B[0][n], vgpr1 = B[1][n]
//            lane n+16      vgpr0 = B[2][n], vgpr1 = B[3][n]
//   C/D    : lane n (0..15) vgpr v = D[v][n] ; lane n+16 vgpr v = D[v+8][n]

template <int K, int OUT>
__global__ void k_gemm_wmma(const float* __restrict__ X,
                            const float* __restrict__ W,
                            const float* __restrict__ dinv,
                            const float* __restrict__ bias,
                            float* __restrict__ H,
                            float* __restrict__ AGG, int nrows) {
    __shared__ float xs[16 * K];
    const int tid  = threadIdx.x;
    const int lane = tid & 31;
    const int wave = tid >> 5;         // column tile id, 0..OUT/16-1
    const int row0 = blockIdx.x * 16;

    // L2-hint prefetch of the (tiny, fully cached) weight panel.
    __builtin_prefetch(W, 0, 3);

    // Stage the 16 x K fp32 A-tile into LDS (coalesced along K).
    for (int i = tid; i < 16 * K; i += blockDim.x) {
        const int r = i / K;
        const int c = i % K;
        const int row = row0 + r;
        xs[i] = (row < nrows) ? X[row * K + c] : 0.0f;
    }
    __syncthreads();

    const int m    = lane & 15;        // row within tile / col within tile
    const int koff = (lane >> 4) * 2;  // half-wave K offset: 0 or 2
    const int col  = wave * 16 + m;

    v8f acc = {};
#pragma unroll 8
    for (int k = 0; k < K; k += 4) {
        const int kk = k + koff;
        v2f a, b;
        a.x = xs[m * K + kk];
        a.y = xs[m * K + kk + 1];
        b.x = W[kk * OUT + col];
        b.y = W[(kk + 1) * OUT + col];
        // 8 args: (neg_a, A, neg_b, B, c_mod, C, reuse_a, reuse_b)
        acc = __builtin_amdgcn_wmma_f32_16x16x4_f32(
            false, a, false, b, (short)0, acc, false, false);
    }

    const int mbase = (lane >> 4) * 8;
    const float bcol = bias[col];

    if (row0 + 16 <= nrows) {
        // Full tile: unguarded, half-wave-coalesced stores.
#pragma unroll
        for (int v = 0; v < 8; ++v) {
            const int row = row0 + mbase + v;
            const float hv = acc[v];
            const float di = dinv[row];
            H[row * OUT + col]   = hv;
            AGG[row * OUT + col] = bcol + di * di * hv;
        }
    } else {
#pragma unroll
        for (int v = 0; v < 8; ++v) {
            const int row = row0 + mbase + v;
            if (row < nrows) {
                const float hv = acc[v];
                const float di = dinv[row];
                H[row * OUT + col]   = hv;
                AGG[row * OUT + col] = bcol + di * di * hv;
            }
        }
    }
}

// --------------------------- relu / edge scatter ---------------------------

__global__ void k_relu(float* a, int n) {
    int i = blockIdx.x * blockDim.x + threadIdx.x;
    if (i < n) a[i] = fmaxf(a[i], 0.0f);
}

// One wave per edge: gather h[src]*norm, atomic scatter-add into agg[dst].
// F=64 -> lane handles features {lane, lane+32}; F=32 -> feature = lane.
template <int F>
__global__ void k_edge_prop(const int* __restrict__ src,
                            const int* __restrict__ dst,
                            const float* __restrict__ dinv,
                            const float* __restrict__ h,
                            float* agg, int e) {
    const int wavesPerBlock = blockDim.x >> 5;
    const int edge = blockIdx.x * wavesPerBlock + (threadIdx.x >> 5);
    if (edge >= e) return;
    const int lane = threadIdx.x & 31;
    const int s = src[edge];
    const int d = dst[edge];
    const float norm = dinv[s] * dinv[d];
#pragma unroll
    for (int f = lane; f < F; f += 32) {
        atomicAdd(&agg[d * F + f], h[s * F + f] * norm);
    }
}

// ------------------------------- launcher ----------------------------------

extern "C" void kernel_launch(void* const* d_in, const int* in_sizes, int n_in,
                              void* d_out, int out_size, void* d_ws, size_t ws_size,
                              hipStream_t stream) {
    const float* x   = (const float*)d_in[0];   // [N,128]
    const int*   ei  = (const int*)  d_in[1];   // [2,E] row-major
    const float* W1  = (const float*)d_in[2];   // [128,64]
    const float* b1  = (const float*)d_in[3];   // [64]
    const float* W2  = (const float*)d_in[4];   // [64,32]
    const float* b2  = (const float*)d_in[5];   // [32]
    float*       out = (float*)d_out;           // [N,32]

    const int N = in_sizes[0] / N_FEAT_IN;
    const int E = in_sizes[1] / 2;
    const int* src = ei;
    const int* dst = ei + E;

    // Workspace carve-up (all fully rewritten each call -> deterministic).
    float* deg  = (float*)d_ws;          // N
    float* dinv = deg  + N;              // N
    float* h1   = dinv + N;              // N*64
    float* agg1 = h1   + (size_t)N * 64; // N*64
    float* h2   = agg1 + (size_t)N * 64; // N*32

    const int T = 256;

    // --- normalization ---
    k_init_deg<<<(N + T - 1) / T, T, 0, stream>>>(deg, N);
    k_count_deg<<<(E + T - 1) / T, T, 0, stream>>>(dst, deg, E);
    k_dinv<<<(N + T - 1) / T, T, 0, stream>>>(deg, dinv, N);

    const int tiles = (N + 15) / 16;
    const int edgeBlocks = (E + 7) / 8;  // 8 waves (edges) per 256-thread block

    // --- layer 1: h1 = x @ W1 (+ fused agg1 init) ; edge scatter ; relu ---
    k_gemm_wmma<128, 64><<<tiles, 128, 0, stream>>>(x, W1, dinv, b1, h1, agg1, N);
    k_edge_prop<64><<<edgeBlocks, T, 0, stream>>>(src, dst, dinv, h1, agg1, E);
    k_relu<<<(N * 64 + T - 1) / T, T, 0, stream>>>(agg1, N * 64);

    // --- layer 2: h2 = relu(agg1) @ W2 (+ fused out init) ; edge scatter ---
    k_gemm_wmma<64, 32><<<tiles, 64, 0, stream>>>(agg1, W2, dinv, b2, h2, out, N);
    k_edge_prop<32><<<edgeBlocks, T, 0, stream>>>(src, dst, dinv, h2, out, E);
}